// Geoneigh_FPT_v2_37855841747283
// MI455X (gfx1250) — compile-verified
//
#include <hip/hip_runtime.h>

// Fused geometric-neighbor attention for MI455X (gfx1250, wave32).
// One wave = one (token n, head h). 16 x v_wmma_f32_16x16x4_f32 per wave
// implement the [16x32]x[32x32] edge linear; LN + softmax-over-K done with
// lane shuffles; single pass over HBM (~304 MB -> ~13us roofline).

typedef __attribute__((ext_vector_type(2))) float v2f;
typedef __attribute__((ext_vector_type(8))) float v8f;

namespace {
constexpr int kH    = 4;     // heads
constexpr int kK    = 16;    // neighbors
constexpr int kD    = 32;    // head dim
constexpr int kWRow = 36;    // padded LDS row stride (floats) -> conflict-free
constexpr float kLnEps = 1e-5f;
constexpr float kInvScale = 5.656854249492381f;  // sqrt(32) = 1/scale
}

__global__ __launch_bounds__(256)
void geoneigh_attn_kernel(const float* __restrict__ q,      // [N,H,D]
                          const float* __restrict__ kn,     // [N,K,H,D]
                          const float* __restrict__ vn,     // [N,K,H,D]
                          const float* __restrict__ geo,    // [N,K,D]
                          const float* __restrict__ Wm,     // [D,D]
                          const float* __restrict__ bias,   // [D]
                          const float* __restrict__ gma,    // [D]
                          const float* __restrict__ bta,    // [D]
                          float* __restrict__ out)          // [N,H,D]
{
    __shared__ float wlds[kD * kWRow];

    const int tid = threadIdx.x;
    // Stage W (4 KB) into LDS once per block, padded rows (36 floats) so the
    // 16-lane row-indexed ds_load_b64 reads hit distinct banks.
    for (int i = tid; i < kD * kD; i += 256)
        wlds[(i >> 5) * kWRow + (i & 31)] = Wm[i];
    __syncthreads();

    const int lane = tid & 31;
    const int m    = lane & 15;   // row (k) for A, column (e mod 16) for B/C/D
    const int hi   = lane >> 4;   // half-wave select
    const int g    = blockIdx.x * 8 + (tid >> 5);   // flat (n,h)
    const int n    = g >> 2;
    const int h    = g & 3;

    // Per-lane column constants (column e = m for tile0, m+16 for tile1).
    const float b0c = bias[m],    b1c = bias[m + 16];
    const float g0c = gma[m],     g1c = gma[m + 16];
    const float be0 = bta[m],     be1 = bta[m + 16];

    // ---- Build A fragments: lane (m,hi) owns edge row k=m, d = 4kk+2hi(+1).
    const float* qrow  = q   + ((size_t)n * kH + h) * kD;
    const float* knrow = kn  + (((size_t)n * kK + m) * kH + h) * kD;
    const float* grow  = geo + ((size_t)n * kK + m) * kD;

    v2f a[8];
#pragma unroll
    for (int kk = 0; kk < 8; ++kk) {
        const int d = 4 * kk + 2 * hi;
        const v2f qv = *(const v2f*)(qrow  + d);
        const v2f kv = *(const v2f*)(knrow + d);
        const v2f gv = *(const v2f*)(grow  + d);
        a[kk] = (qv - kv + gv) * kInvScale;
    }

    // ---- Edge linear: [16x32] @ W^T -> two 16x16 f32 accumulators.
    v8f acc0 = {};   // output cols e = 0..15
    v8f acc1 = {};   // output cols e = 16..31
#pragma unroll
    for (int kk = 0; kk < 8; ++kk) {
        const int c = 4 * kk + 2 * hi;
        const v2f bf0 = *(const v2f*)(&wlds[m * kWRow + c]);          // W[m][c..c+1]
        const v2f bf1 = *(const v2f*)(&wlds[(m + 16) * kWRow + c]);   // W[m+16][c..c+1]
        acc0 = __builtin_amdgcn_wmma_f32_16x16x4_f32(false, a[kk], false, bf0,
                                                     (short)0, acc0, false, false);
        acc1 = __builtin_amdgcn_wmma_f32_16x16x4_f32(false, a[kk], false, bf1,
                                                     (short)0, acc1, false, false);
    }

    // ---- bias + ReLU + LayerNorm over e (row k = v + 8*hi per acc VGPR v).
    float wn0[8], wn1[8];
#pragma unroll
    for (int v = 0; v < 8; ++v) {
        const float y0 = fmaxf(acc0[v] + b0c, 0.0f);
        const float y1 = fmaxf(acc1[v] + b1c, 0.0f);
        float s  = y0 + y1;
        float sq = y0 * y0 + y1 * y1;
#pragma unroll
        for (int msk = 1; msk < 16; msk <<= 1) {   // reduce across 16 cols/half
            s  += __shfl_xor(s,  msk, 32);
            sq += __shfl_xor(sq, msk, 32);
        }
        const float mu  = s * (1.0f / 32.0f);
        const float var = sq * (1.0f / 32.0f) - mu * mu;
        const float rs  = __builtin_amdgcn_rsqf(var + kLnEps);
        wn0[v] = (y0 - mu) * rs * g0c + be0;
        wn1[v] = (y1 - mu) * rs * g1c + be1;
    }

    // ---- softmax over K=16 per column e (8 rows in-lane + partner half).
    float mx0 = wn0[0], mx1 = wn1[0];
#pragma unroll
    for (int v = 1; v < 8; ++v) {
        mx0 = fmaxf(mx0, wn0[v]);
        mx1 = fmaxf(mx1, wn1[v]);
    }
    mx0 = fmaxf(mx0, __shfl_xor(mx0, 16, 32));
    mx1 = fmaxf(mx1, __shfl_xor(mx1, 16, 32));

    float p0[8], p1[8], den0 = 0.0f, den1 = 0.0f;
#pragma unroll
    for (int v = 0; v < 8; ++v) {
        p0[v] = __expf(wn0[v] - mx0);  den0 += p0[v];
        p1[v] = __expf(wn1[v] - mx1);  den1 += p1[v];
    }
    den0 += __shfl_xor(den0, 16, 32);
    den1 += __shfl_xor(den1, 16, 32);
    const float r0 = 1.0f / den0;
    const float r1 = 1.0f / den1;

    // ---- attn-weighted sum of v_neigh (row k = v + 8*hi, col e = m / m+16).
    const float* vbase = vn + ((size_t)n * kK * kH + h) * kD;
    float o0 = 0.0f, o1 = 0.0f;
#pragma unroll
    for (int v = 0; v < 8; ++v) {
        const float* vrow = vbase + (size_t)(v + 8 * hi) * kH * kD;
        o0 += p0[v] * vrow[m];
        o1 += p1[v] * vrow[m + 16];
    }
    o0 *= r0;
    o1 *= r1;
    o0 += __shfl_xor(o0, 16, 32);   // combine the two k-halves
    o1 += __shfl_xor(o1, 16, 32);

    // lanes 0-15 own e=m (o0), lanes 16-31 own e=m+16 (o1): contiguous 128B.
    out[((size_t)n * kH + h) * kD + lane] = hi ? o1 : o0;
}

extern "C" void kernel_launch(void* const* d_in, const int* in_sizes, int n_in,
                              void* d_out, int out_size, void* d_ws, size_t ws_size,
                              hipStream_t stream) {
    (void)in_sizes; (void)n_in; (void)out_size; (void)d_ws; (void)ws_size;
    const float* q    = (const float*)d_in[0];
    const float* kn   = (const float*)d_in[1];
    const float* vn   = (const float*)d_in[2];
    const float* geo  = (const float*)d_in[3];
    const float* Wm   = (const float*)d_in[4];
    const float* bias = (const float*)d_in[5];
    const float* gma  = (const float*)d_in[6];
    const float* bta  = (const float*)d_in[7];
    float* out = (float*)d_out;

    const int nTok = 16384;
    const int grid = (nTok * kH) / 8;   // 8 waves (n,h) pairs per 256-thread block
    geoneigh_attn_kernel<<<grid, 256, 0, stream>>>(q, kn, vn, geo, Wm, bias,
                                                   gma, bta, out);
}